// GaussianSampler_36197984371157
// MI455X (gfx1250) — compile-verified
//
#include <hip/hip_runtime.h>
#include <hip/hip_bf16.h>

typedef float v2f __attribute__((ext_vector_type(2)));
typedef float v4f __attribute__((ext_vector_type(4)));
typedef float v8f __attribute__((ext_vector_type(8)));

#define N_DIM 512
#define ROWS  256            // B*S = 2*128
#define TILE  16

// ---------------------------------------------------------------------------
// Kernel 1: zero-fill std_mat (256 MB, write-once, mostly-zero diag matrix).
// Non-temporal 128-bit stores: the buffer exceeds the 192MB L2, stream it.
// ---------------------------------------------------------------------------
__global__ void __launch_bounds__(256) zero_stdmat_kernel(v4f* __restrict__ p) {
    size_t i = (size_t)blockIdx.x * blockDim.x + threadIdx.x;
    v4f z = {0.f, 0.f, 0.f, 0.f};
    __builtin_nontemporal_store(z, p + i);
}

// ---------------------------------------------------------------------------
// Kernel 2: fused GEMM (fp32 WMMA) + softplus + sample + diagonal scatter.
// One wave (32 lanes) per (rowTile, colTile): rowTile in [0,16), colTile in
// [0,32). Each wave computes two 16x16 tiles of stats sharing the A operand:
//   accv = x_tile @ W[n,     :]^T   (pre-softplus variance logits)
//   accm = x_tile @ W[512+n, :]^T   (mu)
// ---------------------------------------------------------------------------
__global__ void __launch_bounds__(256)
gauss_sampler_kernel(const float* __restrict__ x,
                     const float* __restrict__ W,
                     const float* __restrict__ bias,
                     const float* __restrict__ eps,
                     float* __restrict__ sample,
                     float* __restrict__ mu_out,
                     float* __restrict__ std_mat) {
    const int lane   = threadIdx.x & 31;
    const int wave   = threadIdx.x >> 5;
    const int waveId = blockIdx.x * 8 + wave;   // 0..511
    const int rowTile = waveId >> 5;            // 0..15 (row blocks of 16 over 256 rows)
    const int colTile = waveId & 31;            // 0..31 (col blocks of 16 over 512 n)

    const int rBase = rowTile * TILE;
    const int nBase = colTile * TILE;

    // A (16x4 f32) layout: lanes 0-15 -> M=lane, VGPR j = K=j; lanes 16-31 -> K=j+2.
    // B (4x16 f32) layout mirrors it: lane%16 = N, VGPR j = K = j + 2*(lane>=16).
    const int mrow  = lane & 15;
    const int khalf = (lane >> 4) * 2;

    const float* xrow = x + (size_t)(rBase + mrow) * N_DIM + khalf;
    const float* wvar = W + (size_t)(nBase + mrow) * N_DIM + khalf;
    const float* wmu  = W + (size_t)(N_DIM + nBase + mrow) * N_DIM + khalf;

    v8f accv = {};
    v8f accm = {};
    for (int k = 0; k < N_DIM; k += 4) {
        v2f a  = *(const v2f*)(xrow + k);
        v2f bv = *(const v2f*)(wvar + k);
        v2f bm = *(const v2f*)(wmu  + k);
        // 8 args: (neg_a, A, neg_b, B, c_mod, C, reuse_a, reuse_b)
        accv = __builtin_amdgcn_wmma_f32_16x16x4_f32(
            false, a, false, bv, (short)0, accv, false, false);
        accm = __builtin_amdgcn_wmma_f32_16x16x4_f32(
            false, a, false, bm, (short)0, accm, false, false);
    }

    // C/D layout: VGPR g holds (M = g + 8*(lane>=16), N = lane%16).
    const int ncol   = lane & 15;
    const int n      = nBase + ncol;
    const int rowOff = (lane >> 4) * 8;
    const float bias_v = bias[n];
    const float bias_m = bias[N_DIM + n];

#pragma unroll
    for (int g = 0; g < 8; ++g) {
        const int r  = rBase + rowOff + g;
        const float sv = accv[g] + bias_v;
        const float mu = accm[g] + bias_m;
        // numerically stable softplus
        const float var = fmaxf(sv, 0.f) + log1pf(expf(-fabsf(sv)));
        const size_t idx = (size_t)r * N_DIM + n;
        const float e = eps[idx];
        sample[idx] = mu + sqrtf(var) * e;
        mu_out[idx] = mu;
        std_mat[idx * (size_t)N_DIM + n] = var;   // diagonal scatter
    }
}

extern "C" void kernel_launch(void* const* d_in, const int* in_sizes, int n_in,
                              void* d_out, int out_size, void* d_ws, size_t ws_size,
                              hipStream_t stream) {
    const float* x    = (const float*)d_in[0];
    const float* W    = (const float*)d_in[1];
    const float* bias = (const float*)d_in[2];
    const float* eps  = (const float*)d_in[3];

    float* sample  = (float*)d_out;                       // 256*512
    float* mu_out  = sample + (size_t)ROWS * N_DIM;       // 256*512
    float* std_mat = mu_out + (size_t)ROWS * N_DIM;       // 256*512*512

    // 1) stream-zero std_mat: 67,108,864 floats = 16,777,216 float4
    const size_t n4 = (size_t)ROWS * N_DIM * N_DIM / 4;
    zero_stdmat_kernel<<<(unsigned)(n4 / 256), 256, 0, stream>>>((v4f*)std_mat);

    // 2) fused WMMA GEMM + epilogue: 512 waves = 64 blocks x 8 waves
    gauss_sampler_kernel<<<64, 256, 0, stream>>>(x, W, bias, eps,
                                                 sample, mu_out, std_mat);
}